// AttnLayer_73821897883847
// MI455X (gfx1250) — compile-verified
//
#include <hip/hip_runtime.h>
#include <hip/hip_bf16.h>

// ---------------------------------------------------------------------------
// Two-stage additive attention, B=16 T=8192 D=512 H=8 TOPK=1024.
// Exact algebraic reductions:
//   sum_e((x@W + b)*v)_e == x·(W v) + b·v   -> stage scores are GEMVs
//   topk(softmax(s)) == topk(s)             -> stage-1 softmax eliminated
// Only remaining GEMM: vals = (memory[b, idx] @ w + b) * v  (16384x512x512 f32)
// via V_WMMA_F32_16X16X4_F32 (full fp32; 8.6 GFLOP is far below the WMMA
// ceiling; pipeline is HBM-bound at ~370 MB => O(20us) at 23.3 TB/s).
// v2: w repacked into K-pair-interleaved float2 layout so every B fragment is
// one global_load_b64; each wave owns a 16x128 strip (8 accumulators) so the
// inner loop is 9 b64 loads + 8 wmma per K-step.
// ---------------------------------------------------------------------------

#define BB 16
#define TT 8192
#define DD 512
#define HH 8
#define KTOP 1024

typedef __attribute__((ext_vector_type(2))) float v2f;
typedef __attribute__((ext_vector_type(8))) float v8f;

// ---------------- kernel 1: u0 = w·v, c0 = b·v, u[h] = ws[h]·vs[h], c[h] ----
__global__ void prep_kernel(const float* __restrict__ w, const float* __restrict__ bias,
                            const float* __restrict__ v, const float* __restrict__ wsm,
                            const float* __restrict__ bs, const float* __restrict__ vs,
                            float* __restrict__ u0, float* __restrict__ c0,
                            float* __restrict__ u, float* __restrict__ c) {
    int tid = threadIdx.x, lane = tid & 31, wave = tid >> 5;
    int job = blockIdx.x * 8 + wave;
    const float *a, *bp; float* out;
    if (job < 512)            { a = w + (size_t)job * DD; bp = v; out = u0 + job; }
    else if (job < 4608)      { int j = job - 512; int h = j >> 9; int d = j & 511;
                                a = wsm + ((size_t)h * DD + d) * DD; bp = vs + h * DD;
                                out = u + h * DD + d; }
    else if (job == 4608)     { a = bias; bp = v; out = c0; }
    else if (job < 4617)      { int h = job - 4609; a = bs + h * DD; bp = vs + h * DD; out = c + h; }
    else return;
    float s = 0.f;
    for (int e = lane; e < DD; e += 32) s += a[e] * bp[e];
    for (int m = 16; m > 0; m >>= 1) s += __shfl_xor(s, m, 32);
    if (lane == 0) *out = s;
}

// ---------------- kernel 1b: repack w for WMMA B fragments ------------------
// wpack[kpair*DD + n] = { w[2*kpair][n], w[2*kpair+1][n] }  (float2, 1 MB)
__global__ void repack_w_kernel(const float* __restrict__ w, float2* __restrict__ wpack) {
    int t = blockIdx.x * 256 + threadIdx.x;        // 131072 threads
    int kpair = t >> 9, n = t & 511;
    float2 p;
    p.x = w[(size_t)(2 * kpair) * DD + n];
    p.y = w[(size_t)(2 * kpair + 1) * DD + n];
    wpack[t] = p;
}

// ---------------- kernel 2: score0[b,t] = memory[b,t]·u0 + c0 (HBM-bound) ---
__global__ void score_kernel(const float* __restrict__ mem, const float* __restrict__ u0,
                             const float* __restrict__ c0, float* __restrict__ scores) {
    int tid = threadIdx.x, lane = tid & 31, wave = tid >> 5;
    long row = (long)blockIdx.x * 8 + wave;               // one wave per row
    const float4* a  = (const float4*)(mem + row * DD);
    const float4* uu = (const float4*)u0;
    float s = 0.f;
#pragma unroll
    for (int i = 0; i < 4; ++i) {
        float4 x = a[lane + 32 * i], y = uu[lane + 32 * i];
        s += x.x * y.x + x.y * y.y + x.z * y.z + x.w * y.w;
    }
    for (int m = 16; m > 0; m >>= 1) s += __shfl_xor(s, m, 32);
    if (lane == 0) scores[row] = s + c0[0];
}

// ---------------- kernel 3: per-batch full bitonic sort -> sorted top-1024 --
__global__ void __launch_bounds__(1024) topk_kernel(const float* __restrict__ scores,
                                                    int* __restrict__ idx) {
    extern __shared__ char smem[];
    float* key = (float*)smem;
    int*   vid = (int*)(smem + TT * sizeof(float));
    int tid = threadIdx.x, b = blockIdx.x;
    for (int i = tid; i < TT; i += 1024) { key[i] = scores[b * TT + i]; vid[i] = i; }
    for (int size = 2; size <= TT; size <<= 1) {
        for (int stride = size >> 1; stride > 0; stride >>= 1) {
            __syncthreads();
            for (int i = tid; i < TT; i += 1024) {
                int j = i ^ stride;
                if (j > i) {
                    float ki = key[i], kj = key[j];
                    int   vi = vid[i], vj = vid[j];
                    // descending by value, ties by lower original index (jax top_k order)
                    bool iFirst    = (ki > kj) || (ki == kj && vi < vj);
                    bool descBlock = ((i & size) == 0);
                    if (iFirst != descBlock) {
                        key[i] = kj; key[j] = ki; vid[i] = vj; vid[j] = vi;
                    }
                }
            }
        }
    }
    __syncthreads();
    for (int i = tid; i < KTOP; i += 1024) idx[b * KTOP + i] = vid[i];
}

// ---------------- kernel 4: vals = (memory[b, idx] @ w + b) * v via f32 WMMA -
// One wave owns a 16x128 strip: 8 v8f accumulators share each A fragment.
// A layout (16x4 f32): lanes 0-15 hold K={0,1}, lanes 16-31 hold K={2,3}.
// B fragments come from wpack: one float2 (global_load_b64) per 16-col tile.
__global__ void __launch_bounds__(256) gemm_vals_kernel(
        const float* __restrict__ mem, const float2* __restrict__ wpack,
        const float* __restrict__ bias, const float* __restrict__ v,
        const int* __restrict__ idx, float* __restrict__ vals) {
    int tid = threadIdx.x, lane = tid & 31, wave = tid >> 5;
    int g = blockIdx.x * 8 + wave;                 // 4096 waves total
    int batch = g >> 8, rem = g & 255;
    int mtile = rem >> 2, nstrip = rem & 3;        // 64 M-tiles x 4 N-strips(128)
    int lhalf = lane >> 4;                         // K-pair select (0 or 1)
    int l16   = lane & 15;
    int r = idx[batch * KTOP + mtile * 16 + l16];  // gathered row
    const float* aptr = mem + ((size_t)batch * TT + r) * DD + 2 * lhalf;
    int nbase = nstrip * 128;
    const float2* wp = wpack + nbase + l16;        // + kpair*DD + j*16
    v8f acc[8] = {};
    for (int k0 = 0; k0 < DD; k0 += 4) {
        float2 af = *(const float2*)(aptr + k0);   // 8B aligned (even offset)
        v2f a; a.x = af.x; a.y = af.y;
        int kpair = (k0 >> 1) + lhalf;             // kb = k0 + 2*lhalf
        const float2* wrow = wp + (size_t)kpair * DD;
#pragma unroll
        for (int j = 0; j < 8; ++j) {
            float2 bf2 = wrow[j * 16];
            v2f bf; bf.x = bf2.x; bf.y = bf2.y;
            acc[j] = __builtin_amdgcn_wmma_f32_16x16x4_f32(
                false, a, false, bf, (short)0, acc[j], false, false);
        }
    }
    // epilogue: (acc + bias) * v ; C/D layout: VGPR rr -> M = rr + 8*lhalf
    float* outbase = vals + ((size_t)batch * KTOP + mtile * 16) * DD + nbase + l16;
#pragma unroll
    for (int j = 0; j < 8; ++j) {
        int n = nbase + j * 16 + l16;
        float bj = bias[n], vj = v[n];
#pragma unroll
        for (int rr = 0; rr < 8; ++rr) {
            int M = rr + 8 * lhalf;
            outbase[(size_t)M * DD + j * 16] = (acc[j][rr] + bj) * vj;
        }
    }
}

// ---------------- kernel 5: s2[b,h,k] = vals[b,k]·u[h] + c[h] (LDS-tiled) ---
__global__ void __launch_bounds__(256) s2_kernel(const float* __restrict__ vals,
                                                 const float* __restrict__ u,
                                                 const float* __restrict__ c,
                                                 float* __restrict__ s2) {
    __shared__ float tile[16 * DD];                // 32 KB: 16 rows of vals
    int tid = threadIdx.x, lane = tid & 31, wave = tid >> 5;
    int b = blockIdx.x >> 6, rt = blockIdx.x & 63; // 64 row-tiles per batch
    const float* src = vals + ((size_t)b * KTOP + rt * 16) * DD;
    for (int i = tid; i < 16 * DD; i += 256) tile[i] = src[i];
    __syncthreads();
    int h = wave;                                  // wave w -> head w
    float ureg[16];
#pragma unroll
    for (int i = 0; i < 16; ++i) ureg[i] = u[h * DD + lane + 32 * i];
    float ch = c[h];
    for (int row = 0; row < 16; ++row) {
        float s = 0.f;
#pragma unroll
        for (int i = 0; i < 16; ++i) s += tile[row * DD + lane + 32 * i] * ureg[i];
        for (int m = 16; m > 0; m >>= 1) s += __shfl_xor(s, m, 32);
        if (lane == 0) s2[((size_t)(b * HH + h)) * KTOP + rt * 16 + row] = s + ch;
    }
}

// ---------------- kernel 6: softmax over k per (b,h) + attns = prob @ vals ---
__global__ void __launch_bounds__(512) softmax_attn_kernel(
        const float* __restrict__ s2, const float* __restrict__ vals,
        float* __restrict__ out) {
    __shared__ float probsh[HH][KTOP];             // 32 KB
    __shared__ float red[16];
    __shared__ float bcast;
    int tid = threadIdx.x, lane = tid & 31, wave = tid >> 5;
    int b = blockIdx.x;
    float* attns = out;                            // [B,H,D]
    float* prob  = out + BB * HH * DD;             // [B,H,KTOP]
    for (int h = 0; h < HH; ++h) {
        const float* sp = s2 + ((size_t)(b * HH + h)) * KTOP;
        float x0 = sp[tid], x1 = sp[tid + 512];
        // max-reduce
        float m = fmaxf(x0, x1);
        for (int s = 16; s > 0; s >>= 1) m = fmaxf(m, __shfl_xor(m, s, 32));
        if (lane == 0) red[wave] = m;
        __syncthreads();
        if (tid == 0) { float mm = red[0]; for (int i = 1; i < 16; ++i) mm = fmaxf(mm, red[i]); bcast = mm; }
        __syncthreads();
        float mx = bcast;
        float e0 = __expf(x0 - mx), e1 = __expf(x1 - mx);
        __syncthreads();
        // sum-reduce
        float s = e0 + e1;
        for (int ms = 16; ms > 0; ms >>= 1) s += __shfl_xor(s, ms, 32);
        if (lane == 0) red[wave] = s;
        __syncthreads();
        if (tid == 0) { float ss = red[0]; for (int i = 1; i < 16; ++i) ss += red[i]; bcast = ss; }
        __syncthreads();
        float inv = 1.0f / bcast;
        float p0 = e0 * inv, p1 = e1 * inv;
        probsh[h][tid] = p0; probsh[h][tid + 512] = p1;
        float* pp = prob + ((size_t)(b * HH + h)) * KTOP;
        pp[tid] = p0; pp[tid + 512] = p1;
        __syncthreads();
    }
    // attns: thread owns column d = tid; read vals[b] exactly once
    float acc[HH] = {};
    const float* vb = vals + (size_t)b * KTOP * DD;
    for (int kk = 0; kk < KTOP; ++kk) {
        float x = vb[(size_t)kk * DD + tid];
#pragma unroll
        for (int h = 0; h < HH; ++h) acc[h] += probsh[h][kk] * x;
    }
#pragma unroll
    for (int h = 0; h < HH; ++h) attns[((size_t)(b * HH + h)) * DD + tid] = acc[h];
}

// ---------------------------------------------------------------------------
extern "C" void kernel_launch(void* const* d_in, const int* in_sizes, int n_in,
                              void* d_out, int out_size, void* d_ws, size_t ws_size,
                              hipStream_t stream) {
    const float* memory = (const float*)d_in[0];
    const float* w      = (const float*)d_in[1];
    const float* bias   = (const float*)d_in[2];
    const float* v      = (const float*)d_in[3];
    const float* wsm    = (const float*)d_in[4];
    const float* bs     = (const float*)d_in[5];
    const float* vs     = (const float*)d_in[6];
    float* out = (float*)d_out;

    char* wp = (char*)d_ws;
    float*  scores = (float*) (wp + 0);            //    524,288 B
    int*    idx    = (int*)   (wp + 524288);       //     65,536 B
    float*  u0     = (float*) (wp + 589824);       //      2,048 B
    float*  c0     = (float*) (wp + 591872);       //        256 B
    float*  u      = (float*) (wp + 592128);       //     16,384 B
    float*  c      = (float*) (wp + 608512);       //        256 B
    float*  s2     = (float*) (wp + 608768);       //    524,288 B
    float2* wpack  = (float2*)(wp + 1133056);      //  1,048,576 B
    float*  vals   = (float*) (wp + 2181632);      // 33,554,432 B (~35.7 MB total)

    prep_kernel<<<578, 256, 0, stream>>>(w, bias, v, wsm, bs, vs, u0, c0, u, c);
    repack_w_kernel<<<512, 256, 0, stream>>>(w, wpack);
    score_kernel<<<(BB * TT) / 8, 256, 0, stream>>>(memory, u0, c0, scores);
    topk_kernel<<<BB, 1024, TT * 8, stream>>>(scores, idx);
    gemm_vals_kernel<<<512, 256, 0, stream>>>(memory, wpack, bias, v, idx, vals);
    s2_kernel<<<BB * 64, 256, 0, stream>>>(vals, u, c, s2);
    softmax_attn_kernel<<<BB, 512, 0, stream>>>(s2, vals, out);
}